// RepulsionLoss_65781719105610
// MI455X (gfx1250) — compile-verified
//
#include <hip/hip_runtime.h>
#include <hip/hip_bf16.h>

// ---------------------------------------------------------------------------
// Repulsion loss: d2 via V_WMMA_F32_16X16X4_F32 (Gram trick), top-8 per row
// kept as uint32 bit patterns (non-negative floats sort identically as u32,
// avoiding NaN-canonicalization VALU ops), software-pipelined LDS B-tile
// prefetch (2 tiles / iteration), LDS-staged merge, deterministic reduction.
// ---------------------------------------------------------------------------

typedef float v2f __attribute__((ext_vector_type(2)));
typedef float v8f __attribute__((ext_vector_type(8)));

constexpr int   NPTS   = 8192;
constexpr int   NBATCH = 4;
constexpr int   KNBR   = 8;
constexpr float RADIUS = 0.07f;
constexpr float HPAR   = 0.03f;
constexpr float ALPHA  = 0.1f;

constexpr int THREADS          = 256;              // 8 waves (wave32)
constexpr int WAVES            = THREADS / 32;
constexpr int ROWS_PER_BLOCK   = WAVES * 16;       // 128 query rows per block
constexpr int BLOCKS_PER_BATCH = NPTS / ROWS_PER_BLOCK;     // 64
constexpr int NBLOCKS          = NBATCH * BLOCKS_PER_BATCH; // 256
constexpr int NPARTIAL         = NBLOCKS * WAVES;           // 2048 wave partials

// merge scratch layout (reuses s_pts storage): [(wave*16 + row)*129 + col*8 + k]
constexpr int MROW = 16 * KNBR + 1;   // 129 dwords -> conflict-free lane stride

// Sentinel: +1e30f bit pattern (positive, sorts above any real d2).
#define SENTINEL 0x7149f2caU

// Branchless sorted ascending insert behind a wave-coherent early-out.
// u32 compare == float compare for non-negative values; no canonicalization.
__device__ __forceinline__ void top8_insert_u32(unsigned int (&list)[KNBR],
                                                unsigned int v)
{
    if (v < list[KNBR - 1]) {
#pragma unroll
        for (int k = 0; k < KNBR; ++k) {
            const unsigned int old = list[k];
            list[k] = min(v, old);
            v       = max(v, old);
        }
    }
}

__global__ __launch_bounds__(THREADS)
void repulsion_knn_kernel(const float* __restrict__ pts,
                          float* __restrict__ partials)
{
    __shared__ float4 s_pts[NPTS];   // (x, y, z, |p|^2)  = 128 KB LDS

    const int  tid  = threadIdx.x;
    const int  wave = tid >> 5;
    const int  lane = tid & 31;
    const int  col  = lane & 15;
    const bool hi   = lane >= 16;

    const int b       = blockIdx.x / BLOCKS_PER_BATCH;
    const int rowBase = (blockIdx.x % BLOCKS_PER_BATCH) * ROWS_PER_BLOCK;
    const float* P    = pts + (size_t)b * NPTS * 3;

    // Cooperative load of the whole batch slice into LDS (one-time 96 KB read).
    for (int p = tid; p < NPTS; p += THREADS) {
        float x = P[p * 3 + 0];
        float y = P[p * 3 + 1];
        float z = P[p * 3 + 2];
        s_pts[p] = make_float4(x, y, z, x * x + y * y + z * z);
    }
    __syncthreads();

    const int i0 = rowBase + wave * 16;

    // A operand (16x4):  A[m] = (x, y, z, |p_m|^2), m = col
    //   lanes 0-15 : VGPR0=K0(x), VGPR1=K1(y) ; lanes 16-31 : VGPR0=K2(z), VGPR1=K3(|p|^2)
    const float4 pa = s_pts[i0 + col];
    v2f a;
    a.x = hi ? pa.z : pa.x;
    a.y = hi ? pa.w : pa.y;

    // Per-lane sorted ascending top-8 lists (u32 bit patterns),
    // 8 rows per lane (row = r + 8*hi).
    unsigned int knn[8][KNBR];
#pragma unroll
    for (int r = 0; r < 8; ++r)
#pragma unroll
        for (int k = 0; k < KNBR; ++k)
            knn[r][k] = SENTINEL;

    // Software-pipelined loop over column tiles: 32 columns (2 WMMAs) per trip,
    // next trip's B tiles prefetched up front (wrap-around keeps it branchless,
    // so the compiler waits only past the in-flight prefetches).
    float4 pb0 = s_pts[col];
    float4 pb1 = s_pts[16 + col];

    for (int j0 = 0; j0 < NPTS; j0 += 32) {
        const int    jn  = (j0 + 32) & (NPTS - 1);
        const float4 nb0 = s_pts[jn + col];
        const float4 nb1 = s_pts[jn + 16 + col];

        // B operand (4x16): column n = col;  B[0..2][n] = -2*p_n, B[3][n] = 1
        v2f b0, b1;
        b0.x = -2.0f * (hi ? pb0.z : pb0.x);
        b0.y = hi ? 1.0f : (-2.0f * pb0.y);
        b1.x = -2.0f * (hi ? pb1.z : pb1.x);
        b1.y = hi ? 1.0f : (-2.0f * pb1.y);

        // g[m][n] = |p_m|^2 - 2 * p_m . p_n  (two independent XDL ops)
        v8f acc0 = {};
        v8f acc1 = {};
        acc0 = __builtin_amdgcn_wmma_f32_16x16x4_f32(
            false, a, false, b0, (short)0, acc0, false, false);
        acc1 = __builtin_amdgcn_wmma_f32_16x16x4_f32(
            false, a, false, b1, (short)0, acc1, false, false);

        const float sq0 = pb0.w;   // |p_n|^2 for this lane's column, tile 0
        const float sq1 = pb1.w;   // tile 1
#pragma unroll
        for (int r = 0; r < 8; ++r)
            top8_insert_u32(knn[r],
                            __float_as_uint(fmaxf(acc0[r] + sq0, 0.0f)));
#pragma unroll
        for (int r = 0; r < 8; ++r)
            top8_insert_u32(knn[r],
                            __float_as_uint(fmaxf(acc1[r] + sq1, 0.0f)));

        pb0 = nb0;
        pb1 = nb1;
    }

    // ---- merge the 16 per-lane lists of each row via LDS (reuse s_pts) ----
    __syncthreads();                         // everyone done reading s_pts
    unsigned int* s_m = (unsigned int*)s_pts; // 66 KB used of 128 KB

#pragma unroll
    for (int r = 0; r < 8; ++r) {
        const int m = r + (hi ? 8 : 0);
        unsigned int* dst = s_m + (wave * 16 + m) * MROW + col * KNBR;
#pragma unroll
        for (int k = 0; k < KNBR; ++k) dst[k] = knn[r][k];
    }
    __syncthreads();

    float partial = 0.0f;
    if (lane < 16) {
        unsigned int best[KNBR];
#pragma unroll
        for (int k = 0; k < KNBR; ++k) best[k] = SENTINEL;

        const unsigned int* src = s_m + (wave * 16 + lane) * MROW;
        for (int c = 0; c < 16; ++c) {
#pragma unroll
            for (int k = 0; k < KNBR; ++k)
                top8_insert_u32(best, src[c * KNBR + k]);
        }

        constexpr float INV_H2 = 1.0f / (HPAR * HPAR);
#pragma unroll
        for (int k = 0; k < KNBR; ++k) {
            const float d2 = fmaxf(__uint_as_float(best[k]), 1e-12f);
            const float dn = sqrtf(d2);
            partial += (RADIUS - dn) * __expf(-d2 * INV_H2);
        }
    }

    // wave32 reduction (lanes 16-31 carry 0)
#pragma unroll
    for (int off = 16; off >= 1; off >>= 1)
        partial += __shfl_down(partial, off, 32);

    if (lane == 0)
        partials[blockIdx.x * WAVES + wave] = partial;
}

// Fixed-order final reduction: bit-deterministic across graph replays.
__global__ __launch_bounds__(256)
void repulsion_reduce_kernel(const float* __restrict__ partials,
                             float* __restrict__ out)
{
    __shared__ float red[256];
    float s = 0.0f;
    for (int i = threadIdx.x; i < NPARTIAL; i += 256) s += partials[i];
    red[threadIdx.x] = s;
    __syncthreads();
#pragma unroll
    for (int step = 128; step >= 1; step >>= 1) {
        if (threadIdx.x < step) red[threadIdx.x] += red[threadIdx.x + step];
        __syncthreads();
    }
    if (threadIdx.x == 0)
        out[0] = red[0] * (ALPHA / (float)(NBATCH * NPTS * KNBR));
}

extern "C" void kernel_launch(void* const* d_in, const int* in_sizes, int n_in,
                              void* d_out, int out_size, void* d_ws, size_t ws_size,
                              hipStream_t stream)
{
    (void)in_sizes; (void)n_in; (void)out_size; (void)ws_size;
    const float* pts = (const float*)d_in[0];     // [4, 8192, 3] f32
    float* out       = (float*)d_out;             // scalar f32
    float* partials  = (float*)d_ws;              // 2048 floats of scratch

    repulsion_knn_kernel<<<NBLOCKS, THREADS, 0, stream>>>(pts, partials);
    repulsion_reduce_kernel<<<1, 256, 0, stream>>>(partials, out);
}